// VoxelWithPointProjection_89361089560983
// MI455X (gfx1250) — compile-verified
//
#include <hip/hip_runtime.h>
#include <stdint.h>

// ---------------------------------------------------------------------------
// VoxelWithPointProjection: out[n,c] = voxel[n,c] + (mask[n] ? img[b[n],c,y[n],x[n]] : 0)
// N=1e6, C=64, img = (4,64,376,1241) f32. Pure bandwidth-bound gather+add.
// Pass 1: NCHW->NHWC transpose into d_ws. Tile DMA'd to LDS by the CDNA5
//         Tensor Data Mover (one D# per block; HW inserts the +1dw row pad).
// Pass 2: fully-coalesced 256B-per-point gather + masked add.
// Fallback: direct strided gather if d_ws can't hold the NHWC image.
// ---------------------------------------------------------------------------

typedef float        v4f __attribute__((ext_vector_type(4)));
typedef unsigned int v4u __attribute__((ext_vector_type(4)));
typedef int          v4i __attribute__((ext_vector_type(4)));
typedef int          v8i __attribute__((ext_vector_type(8)));

#define BDIM 4
#define CDIM 64
#define HDIM 376
#define WDIM 1241
#define HW   ((size_t)HDIM * (size_t)WDIM)
#define YT   4                      // y-rows per transpose block

#if defined(__has_builtin)
#if __has_builtin(__builtin_amdgcn_tensor_load_to_lds) && \
    __has_builtin(__builtin_amdgcn_s_wait_tensorcnt)
#define USE_TDM 1
#endif
#endif

// CDNA5 async global->LDS copy (ASYNCcnt-tracked) — fallback path.
__device__ __forceinline__ void async_ld_b32(uint32_t lds_off, const float* gp) {
  asm volatile("global_load_async_to_lds_b32 %0, %1, off"
               :: "v"(lds_off), "v"(gp)
               : "memory");
}
__device__ __forceinline__ void wait_asynccnt0() {
  asm volatile("s_wait_asynccnt 0" ::: "memory");
}

#ifdef USE_TDM
// ---------------------------------------------------------------------------
// TDM transpose: img (B,C,H,W) -> ws (B,H,W,C).
// One block per (b, 4-row y chunk, 32-wide x tile). Wave 0 issues a single
// 3D tensor descriptor: dim0 = 32 x (contiguous), dim1 = 64 c (stride H*W),
// dim2 = 4 y (stride W). pad_interval=32dw/pad_amount=1dw -> LDS row stride
// 33 dwords (conflict-free column reads). OOB x returns zero via tensor_dim0.
// This toolchain exposes the 6-arg builtin (extra int32x8 operand, zeroed).
// ---------------------------------------------------------------------------
__global__ void __launch_bounds__(256)
nchw_to_nhwc_tdm(const float* __restrict__ img, float* __restrict__ ws) {
  __shared__ float tile[YT][CDIM][33];
  const int b   = blockIdx.z;
  const int y0  = blockIdx.y * YT;
  const int x0  = blockIdx.x * 32;
  const int tid = threadIdx.x;

  if (tid < 32) {                                   // wave 0 only: TDM ignores EXEC
    const uint64_t gaddr =
        (uint64_t)(uintptr_t)(img + (((size_t)b * CDIM) * HDIM + y0) * WDIM + x0);
    const uint32_t lds   = (uint32_t)(uintptr_t)&tile[0][0][0];
    const uint32_t td0   = (uint32_t)(WDIM - x0);   // remaining width -> OOB reads 0

    v4u g0;
    g0[0] = 1u;                                     // count=1 (valid user D#)
    g0[1] = lds;                                    // lds_addr (bytes)
    g0[2] = (uint32_t)(gaddr & 0xffffffffu);        // global_addr[31:0]
    g0[3] = (uint32_t)((gaddr >> 32) & 0x01ffffffu) // global_addr[56:32]
          | (2u << 30);                             // type=2 ("image")

    v8i g1;
    g1[0] = (int)((2u << 16)      // data_size = 4 bytes
                | (1u << 20)      // pad_enable
                | (4u << 22));    // pad_interval: 1<<4 * 8B = 32 dwords
                                  // pad_amount = 0 -> 1 dword
    g1[1] = (int)(td0 << 16);                       // tensor_dim0[15:0]
    g1[2] = (int)((td0 >> 16) | ((uint32_t)CDIM << 16)); // td0 hi | tensor_dim1 lo
    g1[3] = (int)(32u << 16);                       // tensor_dim1 hi=0 | tile_dim0=32
    g1[4] = (int)((uint32_t)CDIM | ((uint32_t)YT << 16)); // tile_dim1=64 | tile_dim2=4
    g1[5] = (int)(uint32_t)HW;                      // tensor_dim0_stride lo (c step)
    g1[6] = (int)((uint32_t)WDIM << 16);            // stride0 hi=0 | tensor_dim1_stride lo (y step)
    g1[7] = 0;                                      // stride1[47:16] = 0

    v4i g2;
    g2[0] = YT;                                     // tensor_dim2
    g2[1] = 0; g2[2] = 0; g2[3] = 0;                // tensor_dim3 / dim2_stride / tile_dim3 unused
    v4i g3; g3[0] = 0; g3[1] = 0; g3[2] = 0; g3[3] = 0;
    v8i g4;                                         // extra operand (clang-23 6-arg form)
    g4[0] = 0; g4[1] = 0; g4[2] = 0; g4[3] = 0;
    g4[4] = 0; g4[5] = 0; g4[6] = 0; g4[7] = 0;

    __builtin_amdgcn_tensor_load_to_lds(g0, g1, g2, g3, g4, 0);
    __builtin_amdgcn_s_wait_tensorcnt(0);
  }
  __syncthreads();

  // Write NHWC: c contiguous across lanes, float4 per thread, 4 y rows.
  const int c4 = (tid & 15) * 4;
  const int xl = tid >> 4;                          // 0..15
  #pragma unroll
  for (int yy = 0; yy < YT; ++yy) {
    #pragma unroll
    for (int i = 0; i < 2; ++i) {
      const int x = x0 + xl + 16 * i;
      if (x < WDIM) {
        v4f v;
        v.x = tile[yy][c4 + 0][xl + 16 * i];
        v.y = tile[yy][c4 + 1][xl + 16 * i];
        v.z = tile[yy][c4 + 2][xl + 16 * i];
        v.w = tile[yy][c4 + 3][xl + 16 * i];
        *(v4f*)(ws + (((size_t)b * HDIM + (y0 + yy)) * WDIM + x) * CDIM + c4) = v;
      }
    }
  }
}
#else
// ---------------------------------------------------------------------------
// Fallback transpose: per-lane async global->LDS b32 copies (round-1 path).
// ---------------------------------------------------------------------------
__global__ void __launch_bounds__(256)
nchw_to_nhwc_async(const float* __restrict__ img, float* __restrict__ ws) {
  __shared__ float tile[CDIM][33];
  const int b   = blockIdx.z;
  const int y   = blockIdx.y;
  const int x0  = blockIdx.x * 32;
  const int tid = threadIdx.x;
  {
    const int xi = tid & 31;
    const int c0 = tid >> 5;
    const int x  = x0 + xi;
    if (x < WDIM) {
      const float* gp = img + ((size_t)(b * CDIM + c0) * HDIM + y) * WDIM + x;
      uint32_t lp = (uint32_t)(uintptr_t)&tile[c0][xi];
      #pragma unroll
      for (int k = 0; k < 8; ++k) {
        async_ld_b32(lp, gp);
        gp += 8 * HW;
        lp += 8u * 33u * sizeof(float);
      }
    }
  }
  wait_asynccnt0();
  __syncthreads();
  const int c4 = (tid & 15) * 4;
  const int xl = tid >> 4;
  #pragma unroll
  for (int i = 0; i < 2; ++i) {
    const int x = x0 + xl + 16 * i;
    if (x < WDIM) {
      v4f v;
      v.x = tile[c4 + 0][xl + 16 * i];
      v.y = tile[c4 + 1][xl + 16 * i];
      v.z = tile[c4 + 2][xl + 16 * i];
      v.w = tile[c4 + 3][xl + 16 * i];
      *(v4f*)(ws + (((size_t)b * HDIM + y) * WDIM + x) * CDIM + c4) = v;
    }
  }
}
#endif

// ---------------------------------------------------------------------------
// Gather + masked add. 16 lanes per point, float4 per lane (256B/point
// contiguous in NHWC mode). Indices loaded once per 16-lane group, broadcast
// via wave32 shuffles. NT hints on the streamed voxel/out traffic.
// ---------------------------------------------------------------------------
template <bool NHWC>
__global__ void __launch_bounds__(256)
fuse_kernel(const float* __restrict__ voxel,
            const float* __restrict__ feat,        // NHWC ? ws : original img
            const int*   __restrict__ bidx,
            const int*   __restrict__ grid,        // (N,2): [:,0]=x, [:,1]=y
            const unsigned char* __restrict__ mask,
            float* __restrict__ out, int n_pts) {
  const int tid  = threadIdx.x;
  const int n    = blockIdx.x * 16 + (tid >> 4);
  const int lane = tid & 31;
  const int src  = lane & 16;                      // group leader lane (0 or 16)

  int bi = 0, gx = 0, gy = 0, mk = 0;
  if ((lane & 15) == 0 && n < n_pts) {
    bi = bidx[n];
    gx = grid[2 * (size_t)n + 0];
    gy = grid[2 * (size_t)n + 1];
    mk = (int)mask[n];
  }
  bi = __shfl(bi, src, 32);
  gx = __shfl(gx, src, 32);
  gy = __shfl(gy, src, 32);
  mk = __shfl(mk, src, 32);
  if (n >= n_pts) return;

  const int c4 = (tid & 15) * 4;
  v4f v = __builtin_nontemporal_load(
      (const v4f*)(voxel + (size_t)n * CDIM + c4));

  if (mk) {
    v4f f;
    if (NHWC) {
      const size_t pix = ((size_t)bi * HDIM + gy) * WDIM + gx;
      f = *(const v4f*)(feat + pix * CDIM + c4);
    } else {
      const float* p = feat + (((size_t)bi * CDIM + c4) * HDIM + gy) * WDIM + gx;
      f.x = p[0];
      f.y = p[HW];
      f.z = p[2 * HW];
      f.w = p[3 * HW];
    }
    v += f;
  }
  __builtin_nontemporal_store(v, (v4f*)(out + (size_t)n * CDIM + c4));
}

// ---------------------------------------------------------------------------
extern "C" void kernel_launch(void* const* d_in, const int* in_sizes, int n_in,
                              void* d_out, int out_size, void* d_ws, size_t ws_size,
                              hipStream_t stream) {
  const float*         voxel = (const float*)d_in[0];          // (N,64) f32
  const float*         img   = (const float*)d_in[1];          // (4,64,376,1241) f32
  const int*           bidx  = (const int*)d_in[2];            // (N,) i32
  const int*           grid  = (const int*)d_in[3];            // (N,2) i32
  const unsigned char* mask  = (const unsigned char*)d_in[4];  // (N,) bool
  float*               out   = (float*)d_out;                  // (N,64) f32

  const int    n_pts     = in_sizes[0] / CDIM;
  const size_t img_bytes = (size_t)in_sizes[1] * sizeof(float);
  dim3 gblk((unsigned)((n_pts + 15) / 16));

  if (ws_size >= img_bytes) {
#ifdef USE_TDM
    dim3 tblk((WDIM + 31) / 32, (HDIM + YT - 1) / YT, BDIM);
    nchw_to_nhwc_tdm<<<tblk, 256, 0, stream>>>(img, (float*)d_ws);
#else
    dim3 tblk((WDIM + 31) / 32, HDIM, BDIM);
    nchw_to_nhwc_async<<<tblk, 256, 0, stream>>>(img, (float*)d_ws);
#endif
    fuse_kernel<true><<<gblk, 256, 0, stream>>>(voxel, (const float*)d_ws,
                                                bidx, grid, mask, out, n_pts);
  } else {
    fuse_kernel<false><<<gblk, 256, 0, stream>>>(voxel, img,
                                                 bidx, grid, mask, out, n_pts);
  }
}